// CrAKNVectorAttention_49813030699157
// MI455X (gfx1250) — compile-verified
//
#include <hip/hip_runtime.h>

// ---------------------------------------------------------------------------
// CrAKN vector attention, MI455X (gfx1250).
//
// Math: w = softmax(h, axis=j); out[i,k] = (sum_j w[i,j,k]) * v[i,k].
// sum_j softmax_j(...) == 1 identically, so out == v == feat @ Wv + bv.
// The whole [N,C,C] pipeline (rel, W1, LN, mish, W2, softmax) cancels.
// Remaining work: one 1024x256x256 fp32 GEMM + bias = 134 MFLOP / ~2.3 MB,
// ~100 ns at 23.3 TB/s -- latency bound. Implemented with V_WMMA_F32_16X16X4_F32
// (fp32 in / fp32 accumulate) so precision matches the fp32 reference exactly.
//
// All dimensions are compile-time constants so the inner loop is a clean
// constant-trip-count WMMA pipeline with no remainder control flow.
// ---------------------------------------------------------------------------

typedef __attribute__((ext_vector_type(2))) float v2f;
typedef __attribute__((ext_vector_type(8))) float v8f;

constexpr int GM = 1024;   // rows (N in the reference)
constexpr int GN = 256;    // cols (C)
constexpr int GK = 256;    // reduction (C)
constexpr int TILES_N = GN / 16;            // 16
constexpr int TILES   = (GM / 16) * TILES_N; // 1024 tiles, one wave each
constexpr int WAVES_PER_BLOCK = 8;          // 256 threads, wave32

__global__ __launch_bounds__(32 * WAVES_PER_BLOCK)
void crakn_v_gemm_wmma(const float* __restrict__ A,    // feat [GM,GK] row-major
                       const float* __restrict__ B,    // Wv   [GK,GN] row-major
                       const float* __restrict__ bias, // bv   [GN]
                       float* __restrict__ D)          // out  [GM,GN]
{
    const int lane = threadIdx.x & 31;
    const int wave = (blockIdx.x * (32 * WAVES_PER_BLOCK) + threadIdx.x) >> 5;
    const int tm = (wave / TILES_N) << 4;  // tile row base (M)
    const int tn = (wave % TILES_N) << 4;  // tile col base (N)
    const int half = lane >> 4;            // 0: lanes 0-15, 1: lanes 16-31
    const int l16  = lane & 15;
    const int arow = tm + l16;             // A-matrix row handled by this lane
    const int bcol = tn + l16;             // B/C/D column handled by this lane

    v8f acc = {};

    // fp32 WMMA, K=4 per issue.
    // A 16x4 layout: lanes 0-15 -> M=lane, {v0,v1} = {K0,K1}; lanes 16-31 -> {K2,K3}.
    //   => per-lane contiguous float2 at A[arow][k + 2*half].
    // B 4x16 layout (mirrored): N = lane&15, K = 2*half + {0,1}.
    const float* ap = A + (size_t)arow * GK + (half << 1);
    const float* bp = B + (size_t)(half << 1) * GN + bcol;
    #pragma unroll 8
    for (int k = 0; k < GK; k += 4) {
        v2f a = *(const v2f*)(ap + k);
        v2f b;
        b.x = bp[(size_t)(k + 0) * GN];
        b.y = bp[(size_t)(k + 1) * GN];
        acc = __builtin_amdgcn_wmma_f32_16x16x4_f32(
            /*neg_a=*/false, a, /*neg_b=*/false, b,
            /*c_mod=*/(short)0, acc, /*reuse_a=*/false, /*reuse_b=*/false);
    }

    // C/D 16x16 f32 layout: VGPR r holds (M = r + 8*half, N = lane&15).
    // Same N for all 8 accumulators in a lane -> one scalar bias per lane.
    const float bvv = bias[bcol];
    #pragma unroll
    for (int r = 0; r < 8; ++r) {
        const int m = tm + r + (half << 3);
        D[(size_t)m * GN + bcol] = acc[r] + bvv;
    }
}

extern "C" void kernel_launch(void* const* d_in, const int* in_sizes, int n_in,
                              void* d_out, int out_size, void* d_ws, size_t ws_size,
                              hipStream_t stream)
{
    // setup_inputs() order:
    // 0:feat 1:Wq 2:bq 3:gq 4:betaq 5:Wk 6:bk 7:gk 8:betak
    // 9:Wv 10:bv 11:W1 12:b1 13:g1 14:beta1 15:W2 16:b2
    const float* feat = (const float*)d_in[0];
    const float* Wv   = (const float*)d_in[9];
    const float* bv   = (const float*)d_in[10];
    float*       out  = (float*)d_out;

    dim3 block(32 * WAVES_PER_BLOCK);            // 256 threads = 8 waves
    dim3 grid(TILES / WAVES_PER_BLOCK);          // 128 blocks

    crakn_v_gemm_wmma<<<grid, block, 0, stream>>>(feat, Wv, bv, out);
}